// NTM_Head_39281770889797
// MI455X (gfx1250) — compile-verified
//
#include <hip/hip_runtime.h>
#include <math.h>

#define B_TOT 8192
#define NN    128
#define MM    64
#define HID   256
#define CTRL  256
#define REPRD 70
#define SEQW  8
#define XW    9     // SEQ_WIDTH + 1
#define LSTR  66    // padded LDS row stride for memory tile (bank-conflict free)
#define HSTR  260   // padded LDS row stride for h / ctrl

typedef float v2f __attribute__((ext_vector_type(2)));
typedef float v8f __attribute__((ext_vector_type(8)));
typedef int   i4v __attribute__((vector_size(16)));   // int4 for async-LDS builtin

__device__ __forceinline__ float softplusf_(float x) {
    return fmaxf(x, 0.f) + log1pf(expf(-fabsf(x)));
}
__device__ __forceinline__ float sigmoidf_(float x) { return 1.f / (1.f + expf(-x)); }
__device__ __forceinline__ float lrelu_(float x)    { return x > 0.f ? x : 0.01f * x; }

// ---------------------------------------------------------------------------
// Kernel 1: controller.  16 batch rows per block.
//   h    = leaky(x @ W1 + b1)                (scalar, K=9)
//   ctrl = leaky(h @ W2 + b2)                (WMMA f32 16x16x4, K=256)
//   rv   = ctrl @ Wr + br ; wv = ctrl @ Ww + bw   (scalar dots, 70 outs each)
// ---------------------------------------------------------------------------
__global__ __launch_bounds__(256)
void ntm_ctrl_kernel(const float* __restrict__ x,
                     const float* __restrict__ W1, const float* __restrict__ b1,
                     const float* __restrict__ W2, const float* __restrict__ b2,
                     const float* __restrict__ Wr, const float* __restrict__ br,
                     const float* __restrict__ Ww, const float* __restrict__ bw,
                     float* __restrict__ rv_out, float* __restrict__ wv_out)
{
    __shared__ float lx[16 * XW];
    __shared__ float lh[16 * HSTR];
    __shared__ float lc[16 * HSTR];

    const int tid = threadIdx.x;
    const int b0  = blockIdx.x * 16;

    for (int i = tid; i < 16 * XW; i += 256) lx[i] = x[b0 * XW + i];
    __syncthreads();

    // Stage 1: h (16 x 256)
    for (int t = tid; t < 16 * HID; t += 256) {
        int r = t >> 8, j = t & 255;
        float acc = b1[j];
#pragma unroll
        for (int i = 0; i < XW; ++i) acc += lx[r * XW + i] * W1[i * HID + j];
        lh[r * HSTR + j] = lrelu_(acc);
    }
    __syncthreads();

    // Stage 2: ctrl via WMMA f32 16x16x4.  One 16-row strip, 16 col tiles,
    // 8 waves -> 2 tiles per wave.  EXEC is all-ones (no divergence).
    const int lane  = tid & 31;
    const int wave  = tid >> 5;
    const int mrow  = lane & 15;        // A: M index; B/C/D: N index
    const int khalf = (lane >> 4) << 1; // 0 or 2 (K sub-offset per lane half)

    for (int ct = wave; ct < 16; ct += 8) {
        v8f acc = {};
        for (int k = 0; k < CTRL; k += 4) {
            v2f a, bb;
            // A 16x4 layout: lane half selects K pair {0,1} / {2,3}
            a[0] = lh[mrow * HSTR + k + khalf];
            a[1] = lh[mrow * HSTR + k + khalf + 1];
            // B 4x16 layout (mirror of A): row K = khalf(+1), col N = lane&15
            bb[0] = W2[(k + khalf)     * CTRL + ct * 16 + mrow];
            bb[1] = W2[(k + khalf + 1) * CTRL + ct * 16 + mrow];
            acc = __builtin_amdgcn_wmma_f32_16x16x4_f32(
                false, a, false, bb, (short)0, acc, false, false);
        }
        // D 16x16 layout: VGPR v -> M = v (lanes 0-15) / v+8 (lanes 16-31)
#pragma unroll
        for (int v = 0; v < 8; ++v) {
            int row = v + ((lane >> 4) << 3);
            int col = ct * 16 + mrow;
            lc[row * HSTR + col] = lrelu_(acc[v] + b2[col]);
        }
    }
    __syncthreads();

    // Stage 3: head projections (16 rows x 140 outputs)
    for (int t = tid; t < 16 * 2 * REPRD; t += 256) {
        int r  = t / (2 * REPRD);
        int o  = t % (2 * REPRD);
        int oo = (o < REPRD) ? o : o - REPRD;
        const float* Wm = (o < REPRD) ? Wr : Ww;
        float acc = (o < REPRD) ? br[oo] : bw[oo];
        for (int k = 0; k < CTRL; ++k) acc += lc[r * HSTR + k] * Wm[k * REPRD + oo];
        if (o < REPRD) rv_out[(size_t)(b0 + r) * REPRD + oo] = acc;
        else           wv_out[(size_t)(b0 + r) * REPRD + oo] = acc;
    }
}

// ---------------------------------------------------------------------------
// NTM addressing: wc=softmax(softplus(beta)*cos); interp; shift; sharpen.
// All 256 threads participate (2 threads per memory row n = tid>>1).
// ---------------------------------------------------------------------------
__device__ void address128(const float* __restrict__ cosv,
                           float beta, float g, float s0, float s1, float s2,
                           float gamma,
                           const float* __restrict__ wprev,   // global, row base
                           float* __restrict__ wgb,           // lds scratch [128]
                           float* __restrict__ outw,          // lds result  [128]
                           int n, int half)
{
    float spb = softplusf_(beta);          // > 0 always
    float mx = -1e30f;
    for (int i = 0; i < NN; ++i) mx = fmaxf(mx, cosv[i]);
    mx *= spb;
    float E = expf(spb * cosv[n] - mx);
    if (half == 0) wgb[n] = E;
    __syncthreads();
    float ssum = 0.f;
    for (int i = 0; i < NN; ++i) ssum += wgb[i];
    float gg  = sigmoidf_(g);
    float wgv = gg * (E / ssum) + (1.f - gg) * wprev[n];
    __syncthreads();                       // done reading E values
    if (half == 0) wgb[n] = wgv;
    __syncthreads();

    float sm = fmaxf(s0, fmaxf(s1, s2));
    float e0 = expf(s0 - sm), e1 = expf(s1 - sm), e2 = expf(s2 - sm);
    float es = e0 + e1 + e2;
    // SHIFTS=(-1,0,1): roll(wg,-1)[n]=wg[n+1], roll(wg,1)[n]=wg[n-1]
    float wt = (e0 / es) * wgb[(n + 1) & (NN - 1)]
             + (e1 / es) * wgb[n]
             + (e2 / es) * wgb[(n - 1) & (NN - 1)];
    float gam = 1.f + softplusf_(gamma);
    float wp  = powf(wt + 1e-16f, gam);
    if (half == 0) outw[n] = wp;
    __syncthreads();
    float tot = 0.f;
    for (int i = 0; i < NN; ++i) tot += outw[i];
    __syncthreads();                       // done reading un-normalized values
    if (half == 0) outw[n] = wp / tot;
    __syncthreads();
}

// ---------------------------------------------------------------------------
// Kernel 2: per-batch-row memory pipeline.  memory[b] (128x64, 32KB) is
// streamed from HBM exactly once, HBM -> LDS via the gfx1250 async-LDS path
// (no VGPR round trip, overlaps the preamble compute); everything downstream
// is fused.
// ---------------------------------------------------------------------------
__global__ __launch_bounds__(256)
void ntm_mem_kernel(const float* __restrict__ memory,
                    const float* __restrict__ wrp,   // w_read_prev
                    const float* __restrict__ wwp,   // w_write_prev
                    const float* __restrict__ Wea, const float* __restrict__ bea,
                    const float* __restrict__ Wout, const float* __restrict__ bout,
                    const float* __restrict__ rv_in, const float* __restrict__ wv_in,
                    float* __restrict__ out)
{
    __shared__ float m_[NN * LSTR];
    __shared__ float wv_[REPRD], rv_[REPRD], twv_[REPRD];
    __shared__ float cosb[NN], wgb[NN], waddr[NN];
    __shared__ float ea_e[MM], ea_a[MM], rvec[MM];
    __shared__ float red4[256];

    const int tid  = threadIdx.x;
    const int b    = blockIdx.x;
    const int n    = tid >> 1;
    const int half = tid & 1;
    const int m0   = half * 32;

    // stage memory row-block into LDS (8 x B128 per lane, coalesced,
    // async LDS-direct path tracked by ASYNCcnt)
    const float* gbase = memory + (size_t)b * NN * MM;
#if __has_builtin(__builtin_amdgcn_global_load_async_to_lds_b128)
    for (int v = tid; v < NN * MM / 4; v += 256) {
        int base = v * 4;
        float* ldst = &m_[(base >> 6) * LSTR + (base & 63)];
        __builtin_amdgcn_global_load_async_to_lds_b128(
            (__attribute__((address_space(1))) i4v*)(gbase + base),
            (__attribute__((address_space(3))) i4v*)ldst,
            0, 0);
    }
#else
    const float4* gm = (const float4*)gbase;
    for (int v = tid; v < NN * MM / 4; v += 256) {
        float4 val = gm[v];
        int base = v * 4;
        float* p = &m_[(base >> 6) * LSTR + (base & 63)];
        p[0] = val.x; p[1] = val.y; p[2] = val.z; p[3] = val.w;
    }
#endif

    // overlap: head vectors + tanh while the async DMA fills LDS
    if (tid < REPRD) {
        float w   = wv_in[(size_t)b * REPRD + tid];
        wv_[tid]  = w;
        twv_[tid] = tanhf(w);
        rv_[tid]  = rv_in[(size_t)b * REPRD + tid];
    }

#if __has_builtin(__builtin_amdgcn_global_load_async_to_lds_b128)
    asm volatile("s_wait_asynccnt 0x0" ::: "memory");
#endif
    __syncthreads();

    // key norms (redundant per-thread, tiny)
    float nk_w = 0.f, nk_r = 0.f;
    for (int i = 0; i < MM; ++i) {
        nk_w += wv_[i] * wv_[i];
        nk_r += rv_[i] * rv_[i];
    }
    nk_w = sqrtf(nk_w); nk_r = sqrtf(nk_r);

    // wcos over old memory: 2 threads per row, partner combine via shfl_xor
    {
        float dot = 0.f, ssq = 0.f;
        for (int j = 0; j < 32; ++j) {
            float v = m_[n * LSTR + m0 + j];
            dot += wv_[m0 + j] * v;
            ssq += v * v;
        }
        dot += __shfl_xor(dot, 1, 32);
        ssq += __shfl_xor(ssq, 1, 32);
        if (half == 0) cosb[n] = dot / (nk_w * sqrtf(ssq) + 1e-8f);
    }
    __syncthreads();

    // ea = concat([wcos, tanh(wv)]) @ Wea + bea ; e=sigmoid, a=tanh
    if (tid < 2 * MM) {
        int j = tid;
        float acc = bea[j];
        for (int i = 0; i < NN; ++i)    acc += cosb[i] * Wea[i * (2 * MM) + j];
        for (int i = 0; i < REPRD; ++i) acc += twv_[i] * Wea[(NN + i) * (2 * MM) + j];
        if (j < MM) ea_e[j] = sigmoidf_(acc);
        else        ea_a[j - MM] = tanhf(acc);
    }
    __syncthreads();

    // w_write
    address128(cosb, wv_[64], wv_[65], wv_[66], wv_[67], wv_[68], wv_[69],
               wwp + (size_t)b * NN, wgb, waddr, n, half);

    // erase/add update (in place) fused with rcos over mem_new
    {
        float ww = waddr[n];
        float dot = 0.f, ssq = 0.f;
        for (int j = 0; j < 32; ++j) {
            int m  = m0 + j;
            int id = n * LSTR + m;
            float nv = m_[id] * (1.f - ww * ea_e[m]) + ww * ea_a[m];
            m_[id] = nv;
            dot += rv_[m] * nv;
            ssq += nv * nv;
        }
        dot += __shfl_xor(dot, 1, 32);
        ssq += __shfl_xor(ssq, 1, 32);
        if (half == 0) cosb[n] = dot / (nk_r * sqrtf(ssq) + 1e-8f);
    }
    __syncthreads();

    // w_read
    address128(cosb, rv_[64], rv_[65], rv_[66], rv_[67], rv_[68], rv_[69],
               wrp + (size_t)b * NN, wgb, waddr, n, half);

    // r[m] = sum_n w_read[n] * mem_new[n][m]  (4 partials per column)
    {
        int col = tid & 63;
        int seg = tid >> 6;
        float part = 0.f;
        for (int k = 0; k < 32; ++k) {
            int row = seg * 32 + k;
            part += waddr[row] * m_[row * LSTR + col];
        }
        red4[seg * 64 + col] = part;
    }
    __syncthreads();
    if (tid < MM) {
        float r = red4[tid] + red4[64 + tid] + red4[128 + tid] + red4[192 + tid];
        rvec[tid] = lrelu_(r);
    }
    __syncthreads();
    if (tid < SEQW) {
        float acc = bout[tid];
        for (int m = 0; m < MM; ++m) acc += rvec[m] * Wout[m * SEQW + tid];
        out[(size_t)b * SEQW + tid] = sigmoidf_(acc);
    }
}

extern "C" void kernel_launch(void* const* d_in, const int* in_sizes, int n_in,
                              void* d_out, int out_size, void* d_ws, size_t ws_size,
                              hipStream_t stream) {
    const float* x    = (const float*)d_in[0];
    const float* mem  = (const float*)d_in[1];
    const float* wrp  = (const float*)d_in[2];
    const float* wwp  = (const float*)d_in[3];
    const float* W1   = (const float*)d_in[4];
    const float* b1   = (const float*)d_in[5];
    const float* W2   = (const float*)d_in[6];
    const float* b2   = (const float*)d_in[7];
    const float* Wr   = (const float*)d_in[8];
    const float* br   = (const float*)d_in[9];
    const float* Ww   = (const float*)d_in[10];
    const float* bw   = (const float*)d_in[11];
    const float* Wea  = (const float*)d_in[12];
    const float* bea  = (const float*)d_in[13];
    const float* Wout = (const float*)d_in[14];
    const float* bout = (const float*)d_in[15];
    float* out = (float*)d_out;

    float* rv = (float*)d_ws;                        // 8192*70 floats
    float* wv = rv + (size_t)B_TOT * REPRD;          // 8192*70 floats

    ntm_ctrl_kernel<<<B_TOT / 16, 256, 0, stream>>>(x, W1, b1, W2, b2,
                                                    Wr, br, Ww, bw, rv, wv);
    ntm_mem_kernel<<<B_TOT, 256, 0, stream>>>(mem, wrp, wwp, Wea, bea,
                                              Wout, bout, rv, wv, out);
}